// HL_HGCNN_31507880084191
// MI455X (gfx1250) — compile-verified
//
#include <hip/hip_runtime.h>

// ============================================================================
// HL_HGCNN on MI455X (gfx1250, wave32).
// Dense GEMMs via V_WMMA_F32_16X16X4_F32 (full fp32 precision, 16x16 C tiles,
// K-step 4). One wave computes a 16x64 output strip (4 WMMA accumulators
// sharing each A fragment). Concats and the node->edge gather-average are
// folded into the GEMM A-operand so no concat buffers are materialized.
// Sparse Laplacian / transfer ops are float atomic scatter-adds (L2 atomics).
// ============================================================================

typedef float v2f __attribute__((ext_vector_type(2)));
typedef float v8f __attribute__((ext_vector_type(8)));

#define N_NODES  10000
#define N_EDGES  80000
#define NNZ_T   320000
#define NNZ_S   640000
#define EPS_BN  1e-5f
#define EPS_DEG 1e-6f

static inline unsigned divup(long a, long b) { return (unsigned)((a + b - 1) / b); }

// ---------------------------------------------------------------------------
// Elementwise / scatter helpers
// ---------------------------------------------------------------------------

// dst[r*ldd + c] = src[r*lds + c]
__global__ void k_copy_mat(float* __restrict__ dst, int ldd,
                           const float* __restrict__ src, int lds,
                           int M, int C) {
  long t = (long)blockIdx.x * blockDim.x + threadIdx.x;
  long total = (long)M * C;
  if (t >= total) return;
  int r = (int)(t / C), c = (int)(t % C);
  dst[(long)r * ldd + c] = src[(long)r * lds + c];
}

// deg scatter: one thread per index entry (both rows of edge_index, 2*E total)
__global__ void k_deg(float* __restrict__ deg, const int* __restrict__ idx, int n) {
  int i = blockIdx.x * blockDim.x + threadIdx.x;
  if (i < n) atomicAdd(&deg[idx[i]], 1.0f);
}

// z[ei1[e]] -= ew[e] * x[ei0[e]]   (z pre-initialized to x  =>  z = x - Lx)
__global__ void k_scatter_L(float* __restrict__ z, int ldz,
                            const float* __restrict__ x, int ldx,
                            const int* __restrict__ ei0, const int* __restrict__ ei1,
                            const float* __restrict__ ew, long nnz, int C) {
  long t = (long)blockIdx.x * blockDim.x + threadIdx.x;
  long total = nnz * C;
  if (t >= total) return;
  long e = t / C;
  int  c = (int)(t % C);
  float v = -ew[e] * x[(long)ei0[e] * ldx + c];
  atomicAdd(&z[(long)ei1[e] * ldz + c], v);
}

// x_s2t scatter: out[src[e]] += xs[e]; out[dst[e]] += xs[e]
__global__ void k_scatter_s2t(float* __restrict__ out, int ldo,
                              const float* __restrict__ xs, int lds,
                              const int* __restrict__ src, const int* __restrict__ dst,
                              int E, int C) {
  long t = (long)blockIdx.x * blockDim.x + threadIdx.x;
  long total = (long)E * C;
  if (t >= total) return;
  int e = (int)(t / C), c = (int)(t % C);
  float v = xs[(long)e * lds + c];
  atomicAdd(&out[(long)src[e] * ldo + c], v);
  atomicAdd(&out[(long)dst[e] * ldo + c], v);
}

// x[r] *= 1 / (deg[r] + eps)
__global__ void k_scale_rows(float* __restrict__ x, int ld,
                             const float* __restrict__ deg, int M, int C) {
  long t = (long)blockIdx.x * blockDim.x + threadIdx.x;
  long total = (long)M * C;
  if (t >= total) return;
  int r = (int)(t / C), c = (int)(t % C);
  x[(long)r * ld + c] *= 1.0f / (deg[r] + EPS_DEG);
}

// Small-K conv (init layers, K<=2): out = x@W0 + z@W1 + b
__global__ void k_init_conv(float* __restrict__ out, int ldo,
                            const float* __restrict__ x, const float* __restrict__ z, int K,
                            const float* __restrict__ W0, const float* __restrict__ W1,
                            const float* __restrict__ b, int M, int N) {
  long t = (long)blockIdx.x * blockDim.x + threadIdx.x;
  long total = (long)M * N;
  if (t >= total) return;
  int r = (int)(t / N), c = (int)(t % N);
  float acc = b[c];
  for (int k = 0; k < K; ++k) {
    acc += x[(long)r * K + k] * W0[(long)k * N + c];
    acc += z[(long)r * K + k] * W1[(long)k * N + c];
  }
  out[(long)r * ldo + c] = acc;
}

// ---------------------------------------------------------------------------
// BatchNorm (training mode, per-column over rows) + ReLU
// ---------------------------------------------------------------------------
__global__ void k_bn_stats(const float* __restrict__ x, int ld, int M,
                           float* __restrict__ mean, float* __restrict__ var) {
  int c = blockIdx.x;
  float s = 0.0f, ss = 0.0f;
  for (int r = threadIdx.x; r < M; r += blockDim.x) {
    float v = x[(long)r * ld + c];
    s += v; ss += v * v;
  }
  __shared__ float sh1[256], sh2[256];
  sh1[threadIdx.x] = s; sh2[threadIdx.x] = ss;
  __syncthreads();
  for (int stride = 128; stride > 0; stride >>= 1) {
    if (threadIdx.x < stride) {
      sh1[threadIdx.x] += sh1[threadIdx.x + stride];
      sh2[threadIdx.x] += sh2[threadIdx.x + stride];
    }
    __syncthreads();
  }
  if (threadIdx.x == 0) {
    float mu = sh1[0] / (float)M;
    mean[c] = mu;
    var[c]  = sh2[0] / (float)M - mu * mu;
  }
}

__global__ void k_bn_relu(const float* __restrict__ x, int ldi,
                          float* __restrict__ y, int ldo,
                          const float* __restrict__ mean, const float* __restrict__ var,
                          const float* __restrict__ g, const float* __restrict__ beta,
                          int M, int C) {
  long t = (long)blockIdx.x * blockDim.x + threadIdx.x;
  long total = (long)M * C;
  if (t >= total) return;
  int r = (int)(t / C), c = (int)(t % C);
  float v = (x[(long)r * ldi + c] - mean[c]) * rsqrtf(var[c] + EPS_BN) * g[c] + beta[c];
  y[(long)r * ldo + c] = fmaxf(v, 0.0f);
}

// ---------------------------------------------------------------------------
// WMMA fp32 GEMM:  out = act( A1[M,K1]@Wa[K1,N] + A2[M,K2]@Wb[K2,N] + bias )
//   - one wave per 16x64 output strip (4 WMMA accumulators reuse the A frag)
//   - if gsrc != nullptr, row m of A2 is 0.5*(A2[gsrc[m]] + A2[gdst[m]])
//   - requires M%16==0, N%64==0, K1%4==0, K2%4==0 (all hold in this model)
// ---------------------------------------------------------------------------
__global__ void __launch_bounds__(256)
k_wmma_gemm2(const float* __restrict__ A1, int lda1, int K1,
             const float* __restrict__ Wa,
             const float* __restrict__ A2, int lda2, int K2,
             const float* __restrict__ Wb,
             const int* __restrict__ gsrc, const int* __restrict__ gdst,
             const float* __restrict__ bias,
             float* __restrict__ out, int ldo,
             int M, int N, int relu) {
  const int wave = threadIdx.x >> 5;
  const int lane = threadIdx.x & 31;
  const int tilesN = N >> 6;                    // strips of 64 columns
  const int tilesM = M >> 4;
  const int tile = blockIdx.x * (blockDim.x >> 5) + wave;
  if (tile >= tilesM * tilesN) return;          // wave-uniform exit (EXEC stays full)
  const int tm = tile / tilesN;
  const int tn = tile % tilesN;

  const int mrow = tm * 16 + (lane & 15);
  const int col0 = tn * 64 + (lane & 15);       // column of this lane in strip j=0
  const int kh   = (lane >> 4) << 1;            // 0 for lanes 0-15, 2 for 16-31

  v8f acc0 = {}, acc1 = {}, acc2 = {}, acc3 = {};

  const float* a1row = A1 + (long)mrow * lda1;
  for (int k = 0; k < K1; k += 4) {
    v2f a; a.x = a1row[k + kh]; a.y = a1row[k + kh + 1];
    const float* wp = Wa + (long)(k + kh) * N + col0;
    v2f b0; b0.x = wp[0];  b0.y = wp[N];
    v2f b1; b1.x = wp[16]; b1.y = wp[N + 16];
    v2f b2; b2.x = wp[32]; b2.y = wp[N + 32];
    v2f b3; b3.x = wp[48]; b3.y = wp[N + 48];
    acc0 = __builtin_amdgcn_wmma_f32_16x16x4_f32(false, a, false, b0, (short)0, acc0, false, false);
    acc1 = __builtin_amdgcn_wmma_f32_16x16x4_f32(false, a, false, b1, (short)0, acc1, false, false);
    acc2 = __builtin_amdgcn_wmma_f32_16x16x4_f32(false, a, false, b2, (short)0, acc2, false, false);
    acc3 = __builtin_amdgcn_wmma_f32_16x16x4_f32(false, a, false, b3, (short)0, acc3, false, false);
  }

  if (K2 > 0) {
    if (gsrc != nullptr) {
      const float* p0 = A2 + (long)gsrc[mrow] * lda2;
      const float* p1 = A2 + (long)gdst[mrow] * lda2;
      for (int k = 0; k < K2; k += 4) {
        v2f a;
        a.x = 0.5f * (p0[k + kh]     + p1[k + kh]);
        a.y = 0.5f * (p0[k + kh + 1] + p1[k + kh + 1]);
        const float* wp = Wb + (long)(k + kh) * N + col0;
        v2f b0; b0.x = wp[0];  b0.y = wp[N];
        v2f b1; b1.x = wp[16]; b1.y = wp[N + 16];
        v2f b2; b2.x = wp[32]; b2.y = wp[N + 32];
        v2f b3; b3.x = wp[48]; b3.y = wp[N + 48];
        acc0 = __builtin_amdgcn_wmma_f32_16x16x4_f32(false, a, false, b0, (short)0, acc0, false, false);
        acc1 = __builtin_amdgcn_wmma_f32_16x16x4_f32(false, a, false, b1, (short)0, acc1, false, false);
        acc2 = __builtin_amdgcn_wmma_f32_16x16x4_f32(false, a, false, b2, (short)0, acc2, false, false);
        acc3 = __builtin_amdgcn_wmma_f32_16x16x4_f32(false, a, false, b3, (short)0, acc3, false, false);
      }
    } else {
      const float* a2row = A2 + (long)mrow * lda2;
      for (int k = 0; k < K2; k += 4) {
        v2f a; a.x = a2row[k + kh]; a.y = a2row[k + kh + 1];
        const float* wp = Wb + (long)(k + kh) * N + col0;
        v2f b0; b0.x = wp[0];  b0.y = wp[N];
        v2f b1; b1.x = wp[16]; b1.y = wp[N + 16];
        v2f b2; b2.x = wp[32]; b2.y = wp[N + 32];
        v2f b3; b3.x = wp[48]; b3.y = wp[N + 48];
        acc0 = __builtin_amdgcn_wmma_f32_16x16x4_f32(false, a, false, b0, (short)0, acc0, false, false);
        acc1 = __builtin_amdgcn_wmma_f32_16x16x4_f32(false, a, false, b1, (short)0, acc1, false, false);
        acc2 = __builtin_amdgcn_wmma_f32_16x16x4_f32(false, a, false, b2, (short)0, acc2, false, false);
        acc3 = __builtin_amdgcn_wmma_f32_16x16x4_f32(false, a, false, b3, (short)0, acc3, false, false);
      }
    }
  }

  // C/D layout: VGPR i -> (M = i + 8*(lane>=16), N = lane&15)
  const int orow = tm * 16 + ((lane >> 4) << 3);
  v8f accs[4] = {acc0, acc1, acc2, acc3};
#pragma unroll
  for (int j = 0; j < 4; ++j) {
    const int c = col0 + j * 16;
    const float bc = bias ? bias[c] : 0.0f;
#pragma unroll
    for (int i = 0; i < 8; ++i) {
      float v = accs[j][i] + bc;
      if (relu) v = fmaxf(v, 0.0f);
      out[(long)(orow + i) * ldo + c] = v;
    }
  }
}

// ---------------------------------------------------------------------------
// Final edge readout: out[e] = b + xs[e]·W[0:K] + 0.5*(xt[src]+xt[dst])·W[K:2K]
// ---------------------------------------------------------------------------
__global__ void k_out(const float* __restrict__ xs, const float* __restrict__ xt,
                      const int* __restrict__ src, const int* __restrict__ dst,
                      const float* __restrict__ W, const float* __restrict__ b,
                      float* __restrict__ out, int E, int K) {
  int e = blockIdx.x * blockDim.x + threadIdx.x;
  if (e >= E) return;
  float acc = b[0];
  const float* xr = xs + (long)e * K;
  for (int k = 0; k < K; ++k) acc += xr[k] * W[k];
  const float* t0 = xt + (long)src[e] * K;
  const float* t1 = xt + (long)dst[e] * K;
  for (int k = 0; k < K; ++k) acc += 0.5f * (t0[k] + t1[k]) * W[K + k];
  out[e] = acc;
}

// ===========================================================================
// Host orchestration
// ===========================================================================
struct ConvP { const float *W0, *W1, *b, *beta, *g; };   // JAX pytree order
struct BlkP  { const float *Ws, *Wt, *bs, *bt; ConvP cs, ct; };

extern "C" void kernel_launch(void* const* d_in, const int* in_sizes, int n_in,
                              void* d_out, int out_size, void* d_ws, size_t ws_size,
                              hipStream_t stream) {
  (void)in_sizes; (void)n_in; (void)out_size; (void)ws_size;

  // ---- inputs (setup_inputs insertion order; params flattened in JAX pytree
  //      order: blocks[i]{Ws,Wt,bs,bt,cs{W0,W1,b,beta,g},ct{...}}, init_s,
  //      init_t, out{W,b}) ----
  const float* x_t  = (const float*)d_in[0];   // [10000, 2]
  const float* x_s  = (const float*)d_in[1];   // [80000, 1]
  const float* ew_t = (const float*)d_in[2];   // [320000]
  const float* ew_s = (const float*)d_in[3];   // [640000]
  int idx = 4;
  auto grab = [&]() { return (const float*)d_in[idx++]; };
  BlkP blk[6];
  for (int i = 0; i < 6; ++i) {
    blk[i].Ws = grab(); blk[i].Wt = grab(); blk[i].bs = grab(); blk[i].bt = grab();
    blk[i].cs = { grab(), grab(), grab(), grab(), grab() };
    blk[i].ct = { grab(), grab(), grab(), grab(), grab() };
  }
  ConvP init_s = { grab(), grab(), grab(), grab(), grab() };
  ConvP init_t = { grab(), grab(), grab(), grab(), grab() };
  const float* outW = grab();
  const float* outB = grab();
  const int* ei_t = (const int*)d_in[idx++];   // [2, 320000]
  const int* ei_s = (const int*)d_in[idx++];   // [2, 640000]
  const int* ei   = (const int*)d_in[idx++];   // [2, 80000]
  const int* src = ei, *dst = ei + N_EDGES;

  // ---- workspace carve (fp32) ----
  float* ws = (float*)d_ws;
  float* deg  = ws; ws += N_NODES;
  float* mean = ws; ws += 256;
  float* var  = ws; ws += 256;
  float* xt0  = ws; ws += (size_t)N_NODES * 960;   // dense-skip node features
  float* xs0  = ws; ws += (size_t)N_EDGES * 960;   // dense-skip edge features
  float* s2t  = ws; ws += (size_t)N_NODES * 704;   // edge->node transfer
  float* xta  = ws; ws += (size_t)N_NODES * 256;   // block node act (Wt gemm out)
  float* xtz  = ws; ws += (size_t)N_NODES * 256;   // z = x - Lx (nodes)
  float* xtp  = ws; ws += (size_t)N_NODES * 256;   // conv pre-act / BN in-place
  float* xsa  = ws; ws += (size_t)N_EDGES * 256;
  float* xsz  = ws; ws += (size_t)N_EDGES * 256;
  float* xsp  = ws; ws += (size_t)N_EDGES * 256;

  const int TB = 256;

  // ---- degree ----
  hipMemsetAsync(deg, 0, N_NODES * sizeof(float), stream);
  k_deg<<<divup(2 * N_EDGES, TB), TB, 0, stream>>>(deg, ei, 2 * N_EDGES);

  // ---- init_t: HL conv (K=2) + BN + ReLU -> xt0[:, 0:64] ----
  k_copy_mat<<<divup((long)N_NODES * 2, TB), TB, 0, stream>>>(xtz, 2, x_t, 2, N_NODES, 2);
  k_scatter_L<<<divup((long)NNZ_T * 2, TB), TB, 0, stream>>>(xtz, 2, x_t, 2,
      ei_t, ei_t + NNZ_T, ew_t, NNZ_T, 2);
  k_init_conv<<<divup((long)N_NODES * 64, TB), TB, 0, stream>>>(xtp, 64, x_t, xtz, 2,
      init_t.W0, init_t.W1, init_t.b, N_NODES, 64);
  k_bn_stats<<<64, TB, 0, stream>>>(xtp, 64, N_NODES, mean, var);
  k_bn_relu<<<divup((long)N_NODES * 64, TB), TB, 0, stream>>>(xtp, 64, xt0, 960,
      mean, var, init_t.g, init_t.beta, N_NODES, 64);

  // ---- init_s: HL conv (K=1) + BN + ReLU -> xs0[:, 0:64] ----
  k_copy_mat<<<divup((long)N_EDGES, TB), TB, 0, stream>>>(xsz, 1, x_s, 1, N_EDGES, 1);
  k_scatter_L<<<divup((long)NNZ_S, TB), TB, 0, stream>>>(xsz, 1, x_s, 1,
      ei_s, ei_s + NNZ_S, ew_s, NNZ_S, 1);
  k_init_conv<<<divup((long)N_EDGES * 64, TB), TB, 0, stream>>>(xsp, 64, x_s, xsz, 1,
      init_s.W0, init_s.W1, init_s.b, N_EDGES, 64);
  k_bn_stats<<<64, TB, 0, stream>>>(xsp, 64, N_EDGES, mean, var);
  k_bn_relu<<<divup((long)N_EDGES * 64, TB), TB, 0, stream>>>(xsp, 64, xs0, 960,
      mean, var, init_s.g, init_s.beta, N_EDGES, 64);

  // ---- blocks ----
  const int dlist[6]  = { 64, 128, 192, 320, 448, 704 };  // current skip width
  const int dvlist[6] = { 64, 64, 128, 128, 256, 256 };   // block output width
  int off = 64;
  for (int bi = 0; bi < 6; ++bi) {
    const int d = dlist[bi], dv = dvlist[bi];
    const BlkP& B = blk[bi];

    // NodeEdgeInt: x_s2t = (|par1| x_s0) / deg
    hipMemsetAsync(s2t, 0, (size_t)N_NODES * d * sizeof(float), stream);
    k_scatter_s2t<<<divup((long)N_EDGES * d, TB), TB, 0, stream>>>(s2t, d, xs0, 960,
        src, dst, N_EDGES, d);
    k_scale_rows<<<divup((long)N_NODES * d, TB), TB, 0, stream>>>(s2t, d, deg, N_NODES, d);

    // x_t = relu([x_t0 | x_s2t] @ Wt + bt)      (WMMA, concat folded into K split)
    {
      long tiles = ((long)N_NODES / 16) * (dv / 64);
      k_wmma_gemm2<<<divup(tiles, 8), 256, 0, stream>>>(
          xt0, 960, d, B.Wt, s2t, d, d, B.Wt + (long)d * dv,
          nullptr, nullptr, B.bt, xta, dv, N_NODES, dv, 1);
    }
    // x_s = relu([x_s0 | 0.5*(x_t0[src]+x_t0[dst])] @ Ws + bs)  (gather folded in)
    {
      long tiles = ((long)N_EDGES / 16) * (dv / 64);
      k_wmma_gemm2<<<divup(tiles, 8), 256, 0, stream>>>(
          xs0, 960, d, B.Ws, xt0, 960, d, B.Ws + (long)d * dv,
          src, dst, B.bs, xsa, dv, N_EDGES, dv, 1);
    }

    // NEConv node: pre = x@W0 + (x - Lx)@W1 + b; BN; ReLU
    k_copy_mat<<<divup((long)N_NODES * dv, TB), TB, 0, stream>>>(xtz, dv, xta, dv, N_NODES, dv);
    k_scatter_L<<<divup((long)NNZ_T * dv, TB), TB, 0, stream>>>(xtz, dv, xta, dv,
        ei_t, ei_t + NNZ_T, ew_t, NNZ_T, dv);
    {
      long tiles = ((long)N_NODES / 16) * (dv / 64);
      k_wmma_gemm2<<<divup(tiles, 8), 256, 0, stream>>>(
          xta, dv, dv, B.ct.W0, xtz, dv, dv, B.ct.W1,
          nullptr, nullptr, B.ct.b, xtp, dv, N_NODES, dv, 0);
    }
    k_bn_stats<<<dv, TB, 0, stream>>>(xtp, dv, N_NODES, mean, var);
    k_bn_relu<<<divup((long)N_NODES * dv, TB), TB, 0, stream>>>(xtp, dv, xtp, dv,
        mean, var, B.ct.g, B.ct.beta, N_NODES, dv);

    // NEConv edge
    k_copy_mat<<<divup((long)N_EDGES * dv, TB), TB, 0, stream>>>(xsz, dv, xsa, dv, N_EDGES, dv);
    k_scatter_L<<<divup((long)NNZ_S * dv, TB), TB, 0, stream>>>(xsz, dv, xsa, dv,
        ei_s, ei_s + NNZ_S, ew_s, NNZ_S, dv);
    {
      long tiles = ((long)N_EDGES / 16) * (dv / 64);
      k_wmma_gemm2<<<divup(tiles, 8), 256, 0, stream>>>(
          xsa, dv, dv, B.cs.W0, xsz, dv, dv, B.cs.W1,
          nullptr, nullptr, B.cs.b, xsp, dv, N_EDGES, dv, 0);
    }
    k_bn_stats<<<dv, TB, 0, stream>>>(xsp, dv, N_EDGES, mean, var);
    k_bn_relu<<<divup((long)N_EDGES * dv, TB), TB, 0, stream>>>(xsp, dv, xsp, dv,
        mean, var, B.cs.g, B.cs.beta, N_EDGES, dv);

    // dense skip append
    k_copy_mat<<<divup((long)N_NODES * dv, TB), TB, 0, stream>>>(xt0 + off, 960, xtp, dv, N_NODES, dv);
    k_copy_mat<<<divup((long)N_EDGES * dv, TB), TB, 0, stream>>>(xs0 + off, 960, xsp, dv, N_EDGES, dv);
    off += dv;
  }

  // ---- final readout: [x_s | 0.5*(x_t[src]+x_t[dst])] @ W + b ----
  k_out<<<divup(N_EDGES, TB), TB, 0, stream>>>(xsp, xtp, src, dst, outW, outB,
      (float*)d_out, N_EDGES, 256);
}